// MultiHeadAttentionPooling_31791347925600
// MI455X (gfx1250) — compile-verified
//
#include <hip/hip_runtime.h>
#include <math.h>

typedef __attribute__((ext_vector_type(2))) float v2f;
typedef __attribute__((ext_vector_type(8))) float v8f;
typedef __attribute__((ext_vector_type(4))) int   v4i;

#define HAS_WMMA_F32X4 __has_builtin(__builtin_amdgcn_wmma_f32_16x16x4_f32)
#define HAS_ASYNC_LDS (__has_builtin(__builtin_amdgcn_global_load_async_to_lds_b128) && \
                       __has_builtin(__builtin_amdgcn_s_wait_asynccnt))

#if HAS_ASYNC_LDS
typedef __attribute__((address_space(1))) v4i gv4i;   // global (device) int4
typedef __attribute__((address_space(3))) v4i lv4i;   // LDS (shared) int4
#endif

static __device__ __forceinline__ v8f zero8() {
    v8f z;
#pragma unroll
    for (int i = 0; i < 8; i++) z[i] = 0.0f;
    return z;
}

// D = A(16x4 f32) * B(4x16 f32) + C(16x16 f32), wave32.
static __device__ __forceinline__ v8f wmma_f32_16x16x4(v2f a, v2f b, v8f c) {
#if HAS_WMMA_F32X4
    return __builtin_amdgcn_wmma_f32_16x16x4_f32(
        /*neg_a=*/false, a, /*neg_b=*/false, b,
        /*c_mod=*/(short)0, c, /*reuse_a=*/false, /*reuse_b=*/false);
#else
    // Shuffle-based emulation matching the ISA VGPR layout (fallback only).
    const int lane   = threadIdx.x & 31;
    const int lane16 = lane & 15;
    const int hi     = lane >> 4;
#pragma unroll
    for (int j = 0; j < 8; j++) {
        int M = j + hi * 8;
        float a0 = __shfl(a.x, M, 32);
        float a1 = __shfl(a.y, M, 32);
        float a2 = __shfl(a.x, M + 16, 32);
        float a3 = __shfl(a.y, M + 16, 32);
        float b0 = __shfl(b.x, lane16, 32);
        float b1 = __shfl(b.y, lane16, 32);
        float b2 = __shfl(b.x, lane16 + 16, 32);
        float b3 = __shfl(b.y, lane16 + 16, 32);
        c[j] += a0 * b0 + a1 * b1 + a2 * b2 + a3 * b3;
    }
    return c;
#endif
}

// Problem constants
#define BB    16
#define PP    8192
#define DD    512
#define HH    4
#define CHUNK 256           // rows of P per workgroup
#define TILE  64            // rows per LDS tile
#define NTILE (CHUNK / TILE)
#define CPB   (PP / CHUNK)  // 32 chunks per batch
#define NWG   (BB * CPB)    // 512 workgroups

// LDS layout (floats), padded strides to dodge bank conflicts
#define XT_S 516            // x tile row stride
#define QS_S 516            // q row stride (5 rows: 4 heads + 1 zero row for pad lanes)
#define WT_S 68             // expweight row stride
#define LDS_FLOATS (64 * XT_S + 5 * QS_S + 64 * 16 + 16 * WT_S + 4 + 4 + 16 + 4)
#define LDS_BYTES  (LDS_FLOATS * 4)

#define SCALE 0.04419417382415922f  // 1/sqrt(512)

__global__ __launch_bounds__(128) void mhap_partial(
    const float* __restrict__ x, const float* __restrict__ q,
    float* __restrict__ wsm, float* __restrict__ wsl, float* __restrict__ wsacc) {
    extern __shared__ float smem[];
    float* xt    = smem;                  // [64][XT_S] x tile
    float* qs    = xt + 64 * XT_S;        // [5][QS_S]  queries (row 4 = zeros)
    float* st    = qs + 5 * QS_S;         // [64][16]   raw scores
    float* wt    = st + 64 * 16;          // [16][WT_S] exp-weights (transposed, padded heads)
    float* smm   = wt + 16 * WT_S;        // [4] running max
    float* smnew = smm + 4;               // [4] tile-updated max
    float* smr   = smnew + 4;             // [16] rescale factors (1.0 for pad heads)
    float* sml   = smr + 16;              // [4] running sum

    const int tid    = threadIdx.x;
    const int lane   = tid & 31;
    const int wv     = tid >> 5;        // 4 waves
    const int lane16 = lane & 15;
    const int hi     = lane >> 4;
    const int koff   = hi * 2;
    const int g      = blockIdx.x;
    const int b      = g >> 5;          // /CPB
    const int p0base = (g & 31) * CHUNK;

    if (tid < 16) smr[tid] = 1.0f;
    if (tid < 4) { smm[tid] = -3.4e38f; sml[tid] = 0.0f; }

    // stage q into LDS (rows 0-3 = heads, row 4 = zeros for pad lanes)
    for (int i = tid; i < 5 * 512; i += 128) {
        int h = i >> 9, c = i & 511;
        qs[h * QS_S + c] = (h < 4) ? q[h * 512 + c] : 0.0f;
    }
    __syncthreads();

    v8f acc[8];
#pragma unroll
    for (int nb = 0; nb < 8; nb++) acc[nb] = zero8();

    const int   nq    = (lane16 < 4) ? lane16 : 4;   // pad lanes read the zero row
    const int   d0    = wv * 128;                    // d range owned by this wave
    const int   rbase = wv * 16;                     // score rows owned by this wave
    const int   arow  = rbase + lane16;

    for (int t = 0; t < NTILE; t++) {
        const int p0 = p0base + t * TILE;
        // ---- stream x tile (64x512 f32 = 128 KB) into LDS ----
        const float* xg = x + ((size_t)b * PP + p0) * DD;
#if HAS_ASYNC_LDS
        // CDNA5 direct global->LDS async path: no VGPR round-trip, ASYNCcnt-tracked.
        for (int i = 0; i < 64; i++) {
            __builtin_amdgcn_global_load_async_to_lds_b128(
                (gv4i*)(xg + (size_t)i * DD + tid * 4),
                (lv4i*)(xt + i * XT_S + tid * 4),
                /*offset=*/0, /*cpol=*/0);
        }
        __builtin_amdgcn_s_wait_asynccnt(0);
#else
        for (int i = 0; i < 64; i++) {
            float4 v = *(const float4*)(xg + (size_t)i * DD + tid * 4);
            *(float4*)(xt + i * XT_S + tid * 4) = v;
        }
#endif
        __syncthreads();

        // ---- Phase 1: scores s[p,h] via WMMA, K=512 in 4 independent chains ----
        v8f c0 = zero8(), c1 = zero8(), c2 = zero8(), c3 = zero8();
        for (int k0 = 0; k0 < 512; k0 += 16) {
            {
                v2f a = *(const v2f*)(xt + arow * XT_S + k0 + 0 + koff);
                v2f bb = *(const v2f*)(qs + nq * QS_S + k0 + 0 + koff);
                c0 = wmma_f32_16x16x4(a, bb, c0);
            }
            {
                v2f a = *(const v2f*)(xt + arow * XT_S + k0 + 4 + koff);
                v2f bb = *(const v2f*)(qs + nq * QS_S + k0 + 4 + koff);
                c1 = wmma_f32_16x16x4(a, bb, c1);
            }
            {
                v2f a = *(const v2f*)(xt + arow * XT_S + k0 + 8 + koff);
                v2f bb = *(const v2f*)(qs + nq * QS_S + k0 + 8 + koff);
                c2 = wmma_f32_16x16x4(a, bb, c2);
            }
            {
                v2f a = *(const v2f*)(xt + arow * XT_S + k0 + 12 + koff);
                v2f bb = *(const v2f*)(qs + nq * QS_S + k0 + 12 + koff);
                c3 = wmma_f32_16x16x4(a, bb, c3);
            }
        }
#pragma unroll
        for (int j = 0; j < 8; j++) {
            float s = (c0[j] + c1[j] + c2[j] + c3[j]) * SCALE;
            st[(rbase + hi * 8 + j) * 16 + lane16] = s;   // D layout: M=j+hi*8, N=lane16
        }
        __syncthreads();

        // ---- online softmax bookkeeping ----
        if (tid < 4) {
            int h = tid;
            float tm = -3.4e38f;
            for (int p = 0; p < 64; p++) tm = fmaxf(tm, st[p * 16 + h]);
            float mo = smm[h];
            float mn = fmaxf(mo, tm);
            smnew[h] = mn;
            smr[h]   = __expf(mo - mn);
            smm[h]   = mn;
        }
        __syncthreads();
        for (int i = tid; i < 1024; i += 128) {   // wt[16][64], pad heads -> 0
            int h = i >> 6, p = i & 63;
            float v = 0.0f;
            if (h < 4) v = __expf(st[p * 16 + h] - smnew[h]);
            wt[h * WT_S + p] = v;
        }
        __syncthreads();
        if (tid < 4) {
            int h = tid;
            float s = 0.0f;
            for (int p = 0; p < 64; p++) s += wt[h * WT_S + p];
            sml[h] = sml[h] * smr[h] + s;
        }

        // ---- Phase 2: rescale accumulators, then acc += w^T * x (WMMA) ----
#pragma unroll
        for (int j = 0; j < 8; j++) {
            float rj = smr[j + hi * 8];
#pragma unroll
            for (int nb = 0; nb < 8; nb++) acc[nb][j] *= rj;
        }
        for (int ks = 0; ks < 64; ks += 4) {
            v2f a = *(const v2f*)(wt + lane16 * WT_S + ks + koff);  // A[m=head][k=p]
#pragma unroll
            for (int nb = 0; nb < 8; nb++) {
                int n = d0 + nb * 16 + lane16;
                v2f bb;
                bb.x = xt[(ks + koff) * XT_S + n];
                bb.y = xt[(ks + koff + 1) * XT_S + n];
                acc[nb] = wmma_f32_16x16x4(a, bb, acc[nb]);
            }
        }
        __syncthreads();   // protect xt/wt before next tile load
    }

    // ---- write chunk partials ----
    if (tid < 4) {
        wsm[g * 4 + tid] = smm[tid];
        wsl[g * 4 + tid] = sml[tid];
    }
    if (hi == 0) {   // lanes 0-15 hold heads 0..7 (valid: 0..3)
#pragma unroll
        for (int j = 0; j < 4; j++) {
#pragma unroll
            for (int nb = 0; nb < 8; nb++) {
                int d = d0 + nb * 16 + lane16;
                wsacc[(size_t)g * 2048 + j * 512 + d] = acc[nb][j];
            }
        }
    }
}

__global__ __launch_bounds__(128) void mhap_combine(
    const float* __restrict__ wsm, const float* __restrict__ wsl,
    const float* __restrict__ wsacc, float* __restrict__ out) {
    const int b   = blockIdx.x;
    const int tid = threadIdx.x;
    __shared__ float coef[CPB * 4];

    if (tid < 4) {
        int h = tid;
        float M = -3.4e38f;
        for (int c = 0; c < CPB; c++) M = fmaxf(M, wsm[(b * CPB + c) * 4 + h]);
        float L = 0.0f;
        for (int c = 0; c < CPB; c++)
            L += __expf(wsm[(b * CPB + c) * 4 + h] - M) * wsl[(b * CPB + c) * 4 + h];
        float inv = 1.0f / (L * 4.0f);   // includes head-mean (1/4)
        for (int c = 0; c < CPB; c++)
            coef[c * 4 + h] = __expf(wsm[(b * CPB + c) * 4 + h] - M) * inv;
    }
    __syncthreads();

    const int d0 = tid * 4;
    float4 o = {0.0f, 0.0f, 0.0f, 0.0f};
    for (int c = 0; c < CPB; c++) {
#pragma unroll
        for (int h = 0; h < 4; h++) {
            float w = coef[c * 4 + h];
            const float4 v =
                *(const float4*)&wsacc[((size_t)(b * CPB + c) * 4 + h) * 512 + d0];
            o.x += w * v.x; o.y += w * v.y; o.z += w * v.z; o.w += w * v.w;
        }
    }
    *(float4*)&out[b * 512 + d0] = o;
}

extern "C" void kernel_launch(void* const* d_in, const int* in_sizes, int n_in,
                              void* d_out, int out_size, void* d_ws, size_t ws_size,
                              hipStream_t stream) {
    const float* x = (const float*)d_in[0];   // (16, 8192, 512) f32
    const float* q = (const float*)d_in[1];   // (4, 1, 512)   f32
    float* out = (float*)d_out;               // (16, 512)     f32

    // workspace: m[NWG][4] | l[NWG][4] | acc[NWG][4][512]  (~4.2 MB)
    float* wsm   = (float*)d_ws;
    float* wsl   = wsm + NWG * 4;
    float* wsacc = wsl + NWG * 4;

    mhap_partial<<<dim3(NWG), dim3(128), LDS_BYTES, stream>>>(x, q, wsm, wsl, wsacc);
    mhap_combine<<<dim3(BB), dim3(128), 0, stream>>>(wsm, wsl, wsacc, out);
}